// encoder_65481071410990
// MI455X (gfx1250) — compile-verified
//
#include <hip/hip_runtime.h>
#include <hip/hip_bf16.h>
#include <math.h>

typedef __attribute__((ext_vector_type(2))) float v2f;
typedef __attribute__((ext_vector_type(8))) float v8f;

#define D_IN  512
#define HID   128
#define KDIM  2000
#define DOUT  3

// ---------------------------------------------------------------------------
// Kernel 1: h = tanh(x @ W1^T + b1)   [1024 x 128], fp32 WMMA 16x16x4
// One wave per 16x16 tile of h. EXEC all ones (blockDim=32, no divergence).
//
// VGPR layouts (CDNA5 ISA 7.12.2):
//   A (16x4 f32): lane = hf*16+idx -> M=idx, VGPR v holds K = hf*2+v
//   B (4x16 f32): lane = hf*16+idx -> N=idx, VGPR v holds K = hf*2+v
//   C/D (16x16 f32): VGPR r, lane -> M = r + hf*8, N = idx
// ---------------------------------------------------------------------------
__global__ __launch_bounds__(32) void fc1_wmma_kernel(
    const float* __restrict__ x,    // [1024, 512] row-major
    const float* __restrict__ W1,   // [128, 512]  row-major
    const float* __restrict__ b1,   // [128]
    float* __restrict__ h)          // [1024, 128] row-major
{
    const int lane = threadIdx.x & 31;
    const int hf   = lane >> 4;     // lane half: 0 or 1
    const int idx  = lane & 15;
    const int n0   = blockIdx.x * 16;   // H tile
    const int m0   = blockIdx.y * 16;   // batch tile

    const float* arow = x  + (size_t)(m0 + idx) * D_IN + hf * 2; // A: row m0+idx
    const float* brow = W1 + (size_t)(n0 + idx) * D_IN + hf * 2; // B: W1^T col n0+idx

    v8f c0 = {}, c1 = {}, c2 = {}, c3 = {};
    #pragma unroll 4
    for (int k0 = 0; k0 < D_IN; k0 += 16) {
        v2f a0 = *(const v2f*)(arow + k0 + 0);
        v2f b0 = *(const v2f*)(brow + k0 + 0);
        v2f a1 = *(const v2f*)(arow + k0 + 4);
        v2f b1v = *(const v2f*)(brow + k0 + 4);
        v2f a2 = *(const v2f*)(arow + k0 + 8);
        v2f b2v = *(const v2f*)(brow + k0 + 8);
        v2f a3 = *(const v2f*)(arow + k0 + 12);
        v2f b3v = *(const v2f*)(brow + k0 + 12);
        c0 = __builtin_amdgcn_wmma_f32_16x16x4_f32(false, a0, false, b0,  (short)0, c0, false, false);
        c1 = __builtin_amdgcn_wmma_f32_16x16x4_f32(false, a1, false, b1v, (short)0, c1, false, false);
        c2 = __builtin_amdgcn_wmma_f32_16x16x4_f32(false, a2, false, b2v, (short)0, c2, false, false);
        c3 = __builtin_amdgcn_wmma_f32_16x16x4_f32(false, a3, false, b3v, (short)0, c3, false, false);
    }
    v8f c = (c0 + c1) + (c2 + c3);

    const float bias = b1[n0 + idx];
    #pragma unroll
    for (int r = 0; r < 8; ++r) {
        const int m = m0 + r + hf * 8;
        h[(size_t)m * HID + n0 + idx] = tanhf(c[r] + bias);
    }
}

// ---------------------------------------------------------------------------
// Kernel 2: fused Kenyon top-5 + fc2.
// Block = 256 threads handles one batch row b:
//   - stage h[b,:] (128 f32) in LDS
//   - each thread owns k = tid, tid+256, ... (< 2000):
//       branchless sorted-5 insertion over the 128 products h[b,h]*Wk[k,h]
//       kc = sum(top5); fold into 3 running dots with W2 rows
//   - deterministic LDS tree reduction of the 3 partial sums, add b2.
// Never materializes y[b,k,h] (would be 1 GB).
// ---------------------------------------------------------------------------
__global__ __launch_bounds__(256) void kenyon_fc2_kernel(
    const float* __restrict__ h,    // [1024, 128]
    const float* __restrict__ Wk,   // [2000, 128] row-major
    const float* __restrict__ W2,   // [3, 2000]   row-major
    const float* __restrict__ b2,   // [3]
    float* __restrict__ out)        // [1024, 3]
{
    const int b   = blockIdx.x;
    const int tid = threadIdx.x;

    __shared__ __align__(16) float hs[HID];
    __shared__ float red0[256];
    __shared__ float red1[256];
    __shared__ float red2[256];

    if (tid < HID) hs[tid] = h[(size_t)b * HID + tid];
    __syncthreads();

    float acc0 = 0.f, acc1 = 0.f, acc2 = 0.f;

    for (int k = tid; k < KDIM; k += 256) {
        const float4* wkrow = (const float4*)(Wk + (size_t)k * HID);
        float t0 = -INFINITY, t1 = -INFINITY, t2 = -INFINITY,
              t3 = -INFINITY, t4 = -INFINITY;
        #pragma unroll 8
        for (int j = 0; j < HID / 4; ++j) {
            const float4 w  = wkrow[j];
            const float4 hh = *(const float4*)(hs + j * 4);
            float p[4] = { w.x * hh.x, w.y * hh.y, w.z * hh.z, w.w * hh.w };
            #pragma unroll
            for (int e = 0; e < 4; ++e) {
                float v = p[e], m;
                // branchless insert into sorted list t0 >= t1 >= ... >= t4
                m  = fminf(t0, v); t0 = fmaxf(t0, v); v = m;
                m  = fminf(t1, v); t1 = fmaxf(t1, v); v = m;
                m  = fminf(t2, v); t2 = fmaxf(t2, v); v = m;
                m  = fminf(t3, v); t3 = fmaxf(t3, v); v = m;
                t4 = fmaxf(t4, v);
            }
        }
        const float kc = ((t0 + t1) + (t2 + t3)) + t4;
        acc0 += kc * W2[0 * KDIM + k];
        acc1 += kc * W2[1 * KDIM + k];
        acc2 += kc * W2[2 * KDIM + k];
    }

    red0[tid] = acc0; red1[tid] = acc1; red2[tid] = acc2;
    __syncthreads();
    #pragma unroll
    for (int s = 128; s > 0; s >>= 1) {
        if (tid < s) {
            red0[tid] += red0[tid + s];
            red1[tid] += red1[tid + s];
            red2[tid] += red2[tid + s];
        }
        __syncthreads();
    }
    if (tid < DOUT) {
        float* r = (tid == 0) ? red0 : (tid == 1) ? red1 : red2;
        out[(size_t)b * DOUT + tid] = r[0] + b2[tid];
    }
}

// ---------------------------------------------------------------------------
extern "C" void kernel_launch(void* const* d_in, const int* in_sizes, int n_in,
                              void* d_out, int out_size, void* d_ws, size_t ws_size,
                              hipStream_t stream) {
    (void)in_sizes; (void)n_in; (void)out_size; (void)ws_size;
    const float* x  = (const float*)d_in[0];   // [1024, 512]
    const float* W1 = (const float*)d_in[1];   // [128, 512]
    const float* b1 = (const float*)d_in[2];   // [128]
    const float* Wk = (const float*)d_in[3];   // [2000, 128]
    const float* W2 = (const float*)d_in[4];   // [3, 2000]
    const float* b2 = (const float*)d_in[5];   // [3]
    float* out = (float*)d_out;                // [1024, 3]
    float* h   = (float*)d_ws;                 // [1024, 128] scratch (512 KB)

    // fc1: 64 batch-tiles x 8 hidden-tiles, one wave per 16x16 tile
    fc1_wmma_kernel<<<dim3(HID / 16, 1024 / 16), 32, 0, stream>>>(x, W1, b1, h);
    // fused Kenyon top-5 + fc2: one block per batch row
    kenyon_fc2_kernel<<<1024, 256, 0, stream>>>(h, Wk, W2, b2, out);
}